// AttentionLayer_74526272520760
// MI455X (gfx1250) — compile-verified
//
#include <hip/hip_runtime.h>

// Problem constants (from reference)
#define B_   16
#define TQ_  2048
#define TK_  1024
#define DDEC 256
#define DENC 256
#define DATT 128
#define WIN  3

typedef float f32x2 __attribute__((ext_vector_type(2)));
typedef float f32x8 __attribute__((ext_vector_type(8)));

// ---------------------------------------------------------------------------
// Kernel A: per (batch, window-slot) fold the projections around the <=3 live
// key columns.  Produces, per (b, j):
//   m[b][j][0..255] = Wq^T * (Wk*keys[b,prev+j]+bk)     (score vector)
//   c[b][j]         = bq   . (Wk*keys[b,prev+j]+bk)     (score bias)
//   u[b][j][0..255] = Wo   * (Wv*values[b,prev+j]+bv)   (output vector)
// Work is negligible (48 blocks); weights stay L2-resident.
// ---------------------------------------------------------------------------
__global__ __launch_bounds__(256)
void fold_proj_kernel(const float* __restrict__ keys,
                      const float* __restrict__ values,
                      const float* __restrict__ Wq, const float* __restrict__ bq,
                      const float* __restrict__ Wk, const float* __restrict__ bk,
                      const float* __restrict__ Wv, const float* __restrict__ bv,
                      const float* __restrict__ Wo,
                      const int*   __restrict__ prev_ptr,
                      float* __restrict__ m_ws, float* __restrict__ c_ws,
                      float* __restrict__ u_ws)
{
    const int b   = blockIdx.x / WIN;
    const int j   = blockIdx.x % WIN;
    const int tid = threadIdx.x;            // 0..255
    const int prev = prev_ptr[0];
    const int col  = prev + j;

    float* mo = m_ws + ((size_t)b*WIN + j)*DDEC;
    float* uo = u_ws + ((size_t)b*WIN + j)*DDEC;

    if (col < 0 || col >= TK_) {            // window slot falls off the tensor
        mo[tid] = 0.0f;
        uo[tid] = 0.0f;
        if (tid == 0) c_ws[b*WIN + j] = 0.0f;
        return;
    }

    __shared__ float krow[DENC];
    __shared__ float vrow[DENC];
    __shared__ float kp[DATT];
    __shared__ float vp[DATT];

    krow[tid] = keys  [((size_t)b*TK_ + col)*DENC + tid];
    vrow[tid] = values[((size_t)b*TK_ + col)*DENC + tid];
    __syncthreads();

    if (tid < DATT) {                       // k projection, 128 threads
        float acc = bk[tid];
        const float* w = Wk + (size_t)tid*DENC;
        for (int d = 0; d < DENC; ++d) acc = fmaf(krow[d], w[d], acc);
        kp[tid] = acc;
    } else {                                // v projection, 128 threads
        const int a = tid - DATT;
        float acc = bv[a];
        const float* w = Wv + (size_t)a*DENC;
        for (int d = 0; d < DENC; ++d) acc = fmaf(vrow[d], w[d], acc);
        vp[a] = acc;
    }
    __syncthreads();

    // m[d] = sum_a Wq[a][d]*kp[a]   (Wq is [128,256] row-major)
    // u[d] = sum_a Wo[d][a]*vp[a]   (Wo is [256,128] row-major)
    float macc = 0.0f, uacc = 0.0f;
    const float* wo = Wo + (size_t)tid*DATT;
    for (int a = 0; a < DATT; ++a) {
        macc = fmaf(Wq[(size_t)a*DDEC + tid], kp[a], macc);
        uacc = fmaf(wo[a],                    vp[a], uacc);
    }
    mo[tid] = macc;
    uo[tid] = uacc;

    if (tid == 0) {
        float cc = 0.0f;
        for (int a = 0; a < DATT; ++a) cc = fmaf(bq[a], kp[a], cc);
        c_ws[b*WIN + j] = cc;
    }
}

// ---------------------------------------------------------------------------
// Kernel B: streaming kernel.  64 threads = 2 waves; each wave owns a 16-row
// query tile.  Scores [16x256]@[256x16(3 live)] via V_WMMA_F32_16X16X4_F32
// with a zero-padded B matrix in LDS (no divergent loads in the WMMA loop),
// then 3-way softmax and fully-vectorized float4 stores of out + attn rows.
// ---------------------------------------------------------------------------
#define ROWS_PER_BLOCK 32
#define QPITCH 260   // row pitch (floats): breaks LDS bank aliasing, 16B-aligned

__global__ __launch_bounds__(64)
void attn_out_kernel(const float* __restrict__ query,
                     const float* __restrict__ m_ws,
                     const float* __restrict__ c_ws,
                     const float* __restrict__ u_ws,
                     const float* __restrict__ bo,
                     const int*   __restrict__ prev_ptr,
                     float* __restrict__ out, float* __restrict__ attn)
{
    const int b    = blockIdx.x / (TQ_ / ROWS_PER_BLOCK);
    const int t0   = (blockIdx.x % (TQ_ / ROWS_PER_BLOCK)) * ROWS_PER_BLOCK;
    const int tid  = threadIdx.x;
    const int wave = tid >> 5;
    const int lane = tid & 31;
    const int prev = prev_ptr[0];

    __shared__ __align__(16) float qtile[ROWS_PER_BLOCK][QPITCH]; // A operand
    __shared__ __align__(16) float mmp[16][QPITCH];               // B operand, rows 3..15 = 0
    __shared__ __align__(16) float uu[WIN][DDEC];
    __shared__ __align__(16) float bo_s[DDEC];
    __shared__ float sc[2][16][4];
    __shared__ float cv[WIN];

    // Stage zero-padded B matrix (16 rows, only 0..2 live) + fold vectors.
    for (int i = tid; i < 16*DDEC; i += 64) {
        const int j = i >> 8, d = i & 255;
        mmp[j][d] = (j < WIN) ? m_ws[((size_t)b*WIN + j)*DDEC + d] : 0.0f;
    }
    for (int i = tid; i < WIN*DDEC; i += 64) {
        const int j = i >> 8, d = i & 255;
        uu[j][d] = u_ws[((size_t)b*WIN + j)*DDEC + d];
    }
    for (int i = tid; i < DDEC; i += 64) bo_s[i] = bo[i];
    if (tid < WIN) cv[tid] = c_ws[b*WIN + tid];

    // Stage 32 query rows, coalesced float4 (also the residual input).
    const float* qbase = query + ((size_t)b*TQ_ + t0)*DDEC;
    for (int i = tid; i < ROWS_PER_BLOCK*DDEC/4; i += 64) {
        const int row = i >> 6;               // 64 float4 per row
        const int c4  = i & 63;
        const float4 v = ((const float4*)(qbase + (size_t)row*DDEC))[c4];
        *(float4*)&qtile[row][c4*4] = v;
    }
    __syncthreads();

    // ---- scores via fp32 WMMA: C[16x16] = A[16x256] * B[256x16] ----
    // A frag: lane(0..15)->M, VGPR0/1 -> K = 2*(lane>>4) + {0,1}
    // B frag (mirrored): lane&15 -> N, same K split; rows N>=3 are zero in LDS.
    f32x8 acc = {0.f,0.f,0.f,0.f,0.f,0.f,0.f,0.f};
    const int aRow = wave*16 + (lane & 15);
    const int kOff = 2*(lane >> 4);
    const int nIdx = lane & 15;
    for (int k = 0; k < DDEC; k += 4) {
        f32x2 afrag, bfrag;
        afrag[0] = qtile[aRow][k + kOff];
        afrag[1] = qtile[aRow][k + kOff + 1];
        bfrag[0] = mmp[nIdx][k + kOff];
        bfrag[1] = mmp[nIdx][k + kOff + 1];
        acc = __builtin_amdgcn_wmma_f32_16x16x4_f32(
                  false, afrag, false, bfrag, (short)0, acc, false, false);
    }
    // C/D layout: VGPR r, lanes0-15 -> M=r, lanes16-31 -> M=r+8, N=lane&15.
    if (nIdx < WIN) {
        const int half = lane >> 4;
        #pragma unroll
        for (int r = 0; r < 8; ++r)
            sc[wave][r + 8*half][nIdx] = acc[r];
    }
    __syncthreads();

    const bool v0 = (prev     >= 0) && (prev     < TK_);
    const bool v1 = (prev + 1 >= 0) && (prev + 1 < TK_);
    const bool v2 = (prev + 2 >= 0) && (prev + 2 < TK_);
    const float c0 = cv[0], c1 = cv[1], c2 = cv[2];
    const float sqrtTk = 32.0f;                // Tk * sqrt(1/Tk), Tk = 1024
    const float rs2    = 0.70710678118654752f; // sqrt(0.5)

    for (int rr = 0; rr < 16; ++rr) {
        const int rloc = wave*16 + rr;
        const int t    = t0 + rloc;

        const float s0 = sc[wave][rr][0] + c0;
        const float s1 = sc[wave][rr][1] + c1;
        const float s2 = sc[wave][rr][2] + c2;
        float mx = -3.402823466e38f;
        if (v0) mx = fmaxf(mx, s0);
        if (v1) mx = fmaxf(mx, s1);
        if (v2) mx = fmaxf(mx, s2);
        const float e0 = v0 ? __expf(s0 - mx) : 0.0f;
        const float e1 = v1 ? __expf(s1 - mx) : 0.0f;
        const float e2 = v2 ? __expf(s2 - mx) : 0.0f;
        const float den = e0 + e1 + e2;
        const float inv = den > 0.0f ? 1.0f/den : 0.0f;
        const float p0 = e0*inv, p1 = e1*inv, p2 = e2*inv;
        const float g0 = p0*sqrtTk, g1 = p1*sqrtTk, g2 = p2*sqrtTk;

        // out row: 256 floats, contiguous float4 across lanes.
        float* orow = out + ((size_t)b*TQ_ + t)*DDEC;
        #pragma unroll
        for (int h = 0; h < 2; ++h) {
            const int d = h*128 + lane*4;
            const float4 qv = *(const float4*)&qtile[rloc][d];
            float4 ov;
            ov.x = (g0*uu[0][d+0] + g1*uu[1][d+0] + g2*uu[2][d+0] + bo_s[d+0] + qv.x)*rs2;
            ov.y = (g0*uu[0][d+1] + g1*uu[1][d+1] + g2*uu[2][d+1] + bo_s[d+1] + qv.y)*rs2;
            ov.z = (g0*uu[0][d+2] + g1*uu[1][d+2] + g2*uu[2][d+2] + bo_s[d+2] + qv.z)*rs2;
            ov.w = (g0*uu[0][d+3] + g1*uu[1][d+3] + g2*uu[2][d+3] + bo_s[d+3] + qv.w)*rs2;
            *(float4*)(orow + d) = ov;
        }

        // attn row: 1024 floats of zeros with <=3 probs patched into the
        // overlapping float4 before the store (no cross-lane RMW hazard).
        float* arow = attn + ((size_t)b*TQ_ + t)*TK_;
        #pragma unroll
        for (int h = 0; h < 8; ++h) {
            const int cbase = h*128 + lane*4;
            float4 av = make_float4(0.0f, 0.0f, 0.0f, 0.0f);
            if (cbase <= prev + (WIN-1) && cbase + 3 >= prev) {
                #pragma unroll
                for (int comp = 0; comp < 4; ++comp) {
                    const int jj = cbase + comp - prev;
                    float val = 0.0f;
                    if      (jj == 0) val = p0;
                    else if (jj == 1) val = p1;
                    else if (jj == 2) val = p2;
                    ((float*)&av)[comp] = val;
                }
            }
            *(float4*)(arow + cbase) = av;
        }
    }
}

// ---------------------------------------------------------------------------
extern "C" void kernel_launch(void* const* d_in, const int* in_sizes, int n_in,
                              void* d_out, int out_size, void* d_ws, size_t ws_size,
                              hipStream_t stream)
{
    const float* query  = (const float*)d_in[0];
    const float* keys   = (const float*)d_in[1];
    const float* values = (const float*)d_in[2];
    const float* Wq     = (const float*)d_in[3];
    const float* bq     = (const float*)d_in[4];
    const float* Wk     = (const float*)d_in[5];
    const float* bk     = (const float*)d_in[6];
    const float* Wv     = (const float*)d_in[7];
    const float* bv     = (const float*)d_in[8];
    const float* Wo     = (const float*)d_in[9];
    const float* bo     = (const float*)d_in[10];
    const int*   prev   = (const int*)  d_in[11];

    float* m_ws = (float*)d_ws;                 // B*3*256
    float* u_ws = m_ws + (size_t)B_*WIN*DDEC;   // B*3*256
    float* c_ws = u_ws + (size_t)B_*WIN*DDEC;   // B*3

    float* out  = (float*)d_out;                       // [B,TQ,256]
    float* attn = out + (size_t)B_*TQ_*DDEC;           // [B,TQ,1024]

    fold_proj_kernel<<<B_*WIN, 256, 0, stream>>>(
        keys, values, Wq, bq, Wk, bk, Wv, bv, Wo, prev, m_ws, c_ws, u_ws);

    attn_out_kernel<<<B_*(TQ_/ROWS_PER_BLOCK), 64, 0, stream>>>(
        query, m_ws, c_ws, u_ws, bo, prev, out, attn);
}